// OptimizedDynamicsModel_8881992368587
// MI455X (gfx1250) — compile-verified
//
#include <hip/hip_runtime.h>
#include <hip/hip_bf16.h>

#define B_       16
#define L_       32
#define BL       512      // B_*L_
#define LATENT_  512
#define HID_     768
#define NLAYERS_ 12
#define ACT_DIM_ 32
#define D_STATE_ 16
#define D_CONV_  4
#define D_INNER_ 1536
#define XPROJ_N  (2*D_STATE_ + D_INNER_)   // 1568

typedef float v2f __attribute__((ext_vector_type(2)));
typedef float v8f __attribute__((ext_vector_type(8)));

// ---------------------------------------------------------------------------
// Generic fp32 WMMA GEMM:  C[M,N] = (R? R : 0) + bias[N] + A[M,K] @ W[N,K]^T
// One wave -> one 16x16 C tile. 8 waves per block along N, grid.y over M tiles.
// Uses V_WMMA_F32_16X16X4_F32 (fp32 matrix pipe; weights streamed once at
// full HBM bandwidth; the workload is weight-bandwidth bound).
// ---------------------------------------------------------------------------
__global__ __launch_bounds__(256) void gemm_f32_wmma(
    const float* __restrict__ A, const float* __restrict__ W,
    const float* __restrict__ bias, const float* __restrict__ R,
    float* __restrict__ C, int M, int N, int K)
{
  const int wave = threadIdx.x >> 5;
  const int lane = threadIdx.x & 31;
  const int kh   = lane >> 4;     // which K-half of the fragment this lane holds
  const int mr   = lane & 15;     // row index (A) / col index (B) within tile

  const int nTiles = N >> 4;
  const int nTile  = blockIdx.x * 8 + wave;
  if (nTile >= nTiles) return;    // wave-uniform exit: EXEC stays all-ones per wave
  const int m0 = blockIdx.y << 4;
  const int n0 = nTile << 4;

  // Accumulator init: element i lives at row m0+i+8*kh, col n0+mr.
  const float bv = bias ? bias[n0 + mr] : 0.0f;
  v8f acc;
#pragma unroll
  for (int i = 0; i < 8; ++i) {
    float r = 0.0f;
    if (R) r = R[(size_t)(m0 + i + 8*kh) * N + (n0 + mr)];
    acc[i] = bv + r;
  }

  // A fragment (16x4): lane<16 -> {K=k,k+1}, lane>=16 -> {K=k+2,k+3}; row = mr.
  // B fragment (4x16): same K split; col = mr, read from W (N x K row-major).
  // Both are a single contiguous float2 per lane -> global_load_b64, coalesced.
  const float* ap = A + (size_t)(m0 + mr) * K + 2*kh;
  const float* wp = W + (size_t)(n0 + mr) * K + 2*kh;
#pragma unroll 4
  for (int k = 0; k < K; k += 4) {
    v2f a = *(const v2f*)(ap + k);
    v2f b = *(const v2f*)(wp + k);
    acc = __builtin_amdgcn_wmma_f32_16x16x4_f32(
        /*neg_a=*/false, a, /*neg_b=*/false, b,
        /*c_mod=*/(short)0, acc, /*reuse_a=*/false, /*reuse_b=*/false);
  }

#pragma unroll
  for (int i = 0; i < 8; ++i)
    C[(size_t)(m0 + i + 8*kh) * N + (n0 + mr)] = acc[i];
}

// ---------------------------------------------------------------------------
// Gather action embeddings: ag[row, j] = action_embed[actions[row], j]
// ---------------------------------------------------------------------------
__global__ __launch_bounds__(256) void gather_actions(
    const int* __restrict__ actions, const float* __restrict__ embed,
    float* __restrict__ ag)
{
  int t = blockIdx.x * 256 + threadIdx.x;      // BL * ACT_DIM_
  if (t >= BL * ACT_DIM_) return;
  int row = t >> 5;                            // ACT_DIM_ == 32
  int j   = t & 31;
  ag[t] = embed[(size_t)actions[row] * ACT_DIM_ + j];
}

// x[b,l,h] += pos[l,h]
__global__ __launch_bounds__(256) void add_pos(
    float* __restrict__ x, const float* __restrict__ pos)
{
  int t = blockIdx.x * 256 + threadIdx.x;      // BL * HID_
  if (t >= BL * HID_) return;
  int h = t % HID_;
  int l = (t / HID_) & (L_ - 1);
  x[t] += pos[l * HID_ + h];
}

// ---------------------------------------------------------------------------
// Depthwise causal conv (D_CONV=4) along L + SiLU.
// xz is (BL, 2*D_INNER): first half = xi. One thread per (b, d) channel.
// ---------------------------------------------------------------------------
__global__ __launch_bounds__(256) void conv_silu(
    const float* __restrict__ xz, const float* __restrict__ cw,
    const float* __restrict__ cb, float* __restrict__ xs)
{
  int t = blockIdx.x * 256 + threadIdx.x;
  if (t >= B_ * D_INNER_) return;
  int b = t / D_INNER_;
  int d = t - b * D_INNER_;

  const float w0 = cw[d*4+0], w1 = cw[d*4+1], w2 = cw[d*4+2], w3 = cw[d*4+3];
  const float bb = cb[d];
  const float* xi  = xz + (size_t)b * L_ * (2*D_INNER_) + d;
  float*       out = xs + (size_t)b * L_ * D_INNER_ + d;

  float x0 = 0.f, x1 = 0.f, x2 = 0.f;
  for (int l = 0; l < L_; ++l) {
    float x3 = xi[(size_t)l * (2*D_INNER_)];
    float v  = w0*x0 + w1*x1 + w2*x2 + w3*x3 + bb;
    out[(size_t)l * D_INNER_] = v / (1.0f + __expf(-v));   // SiLU
    x0 = x1; x1 = x2; x2 = x3;
  }
}

// ---------------------------------------------------------------------------
// Selective-state scan. One thread per (b, d); h[16], A[16], dt_w row in regs.
// dbu (BL, 1568): [0:16)=d_in, [16:32)=Bm, [32:)=u.  z from xz second half.
// ---------------------------------------------------------------------------
__global__ __launch_bounds__(256) void ssm_scan(
    const float* __restrict__ dbu, const float* __restrict__ xz,
    const float* __restrict__ dw, const float* __restrict__ db,
    const float* __restrict__ alog, const float* __restrict__ Dp,
    float* __restrict__ y)
{
  int t = blockIdx.x * 256 + threadIdx.x;
  if (t >= B_ * D_INNER_) return;
  int b = t / D_INNER_;
  int d = t - b * D_INNER_;

  float dwr[D_STATE_], Am[D_STATE_], h[D_STATE_];
#pragma unroll
  for (int n = 0; n < D_STATE_; ++n) {
    dwr[n] = dw[d * D_STATE_ + n];
    Am[n]  = -__expf(alog[d * D_STATE_ + n]);
    h[n]   = 0.f;
  }
  const float dbv = db[d];
  const float Dv  = Dp[d];

  for (int l = 0; l < L_; ++l) {
    const float* row = dbu + (size_t)(b * L_ + l) * XPROJ_N;
    float acc = dbv;
#pragma unroll
    for (int n = 0; n < D_STATE_; ++n) acc += row[n] * dwr[n];
    float delta = (acc > 20.f) ? acc : log1pf(__expf(acc));   // softplus
    float u  = row[2*D_STATE_ + d];
    float ys = 0.f;
#pragma unroll
    for (int n = 0; n < D_STATE_; ++n) {
      float bu = delta * row[D_STATE_ + n] * u;
      h[n] = __expf(delta * Am[n]) * h[n] + bu;
      ys += h[n];
    }
    ys += u * Dv;
    float z = xz[(size_t)(b * L_ + l) * (2*D_INNER_) + D_INNER_ + d];
    y[(size_t)(b * L_ + l) * D_INNER_ + d] = ys * (z / (1.f + __expf(-z)));
  }
}

// ---------------------------------------------------------------------------
// LayerNorm over HID_: one block (256 threads = 8 waves) per row.
// ---------------------------------------------------------------------------
__global__ __launch_bounds__(256) void layernorm(
    const float* __restrict__ x, const float* __restrict__ g,
    const float* __restrict__ bta, float* __restrict__ o)
{
  __shared__ float rs[8], rq[8];
  const int row  = blockIdx.x;
  const int lane = threadIdx.x & 31;
  const int wv   = threadIdx.x >> 5;
  const float* xr = x + (size_t)row * HID_;

  float s = 0.f, q = 0.f;
  for (int i = threadIdx.x; i < HID_; i += 256) {
    float v = xr[i];
    s += v; q += v * v;
  }
#pragma unroll
  for (int off = 16; off > 0; off >>= 1) {
    s += __shfl_down(s, off, 32);
    q += __shfl_down(q, off, 32);
  }
  if (lane == 0) { rs[wv] = s; rq[wv] = q; }
  __syncthreads();
  if (wv == 0) {
    float sv = (lane < 8) ? rs[lane] : 0.f;
    float qv = (lane < 8) ? rq[lane] : 0.f;
#pragma unroll
    for (int off = 4; off > 0; off >>= 1) {
      sv += __shfl_down(sv, off, 32);
      qv += __shfl_down(qv, off, 32);
    }
    if (lane == 0) { rs[0] = sv; rq[0] = qv; }
  }
  __syncthreads();
  const float mean = rs[0] * (1.0f / HID_);
  const float var  = rq[0] * (1.0f / HID_) - mean * mean;
  const float rstd = rsqrtf(var + 1e-5f);
  for (int i = threadIdx.x; i < HID_; i += 256)
    o[(size_t)row * HID_ + i] = (xr[i] - mean) * rstd * g[i] + bta[i];
}

// ---------------------------------------------------------------------------
extern "C" void kernel_launch(void* const* d_in, const int* in_sizes, int n_in,
                              void* d_out, int out_size, void* d_ws, size_t ws_size,
                              hipStream_t stream) {
  const float* latents   = (const float*)d_in[0];
  const int*   actions   = (const int*)  d_in[1];
  const float* lpw       = (const float*)d_in[2];
  const float* lpb       = (const float*)d_in[3];
  const float* aemb      = (const float*)d_in[4];
  const float* apw       = (const float*)d_in[5];
  const float* apb       = (const float*)d_in[6];
  const float* pos       = (const float*)d_in[7];
  const float* in_w      = (const float*)d_in[8];
  const float* in_b      = (const float*)d_in[9];
  const float* conv_w    = (const float*)d_in[10];
  const float* conv_b    = (const float*)d_in[11];
  const float* xproj_w   = (const float*)d_in[12];
  const float* xproj_b   = (const float*)d_in[13];
  const float* dt_w      = (const float*)d_in[14];
  const float* dt_b      = (const float*)d_in[15];
  const float* A_log     = (const float*)d_in[16];
  const float* Dp        = (const float*)d_in[17];
  const float* blk_out_w = (const float*)d_in[18];
  const float* blk_out_b = (const float*)d_in[19];
  const float* norm_g    = (const float*)d_in[20];
  const float* norm_b    = (const float*)d_in[21];
  const float* out_w     = (const float*)d_in[22];
  const float* out_b     = (const float*)d_in[23];
  float* out = (float*)d_out;

  // Workspace layout (floats)
  float* x   = (float*)d_ws;                 // BL*HID
  float* xz  = x   + (size_t)BL * HID_;      // BL*2*D_INNER
  float* xs  = xz  + (size_t)BL * 2*D_INNER_;// BL*D_INNER
  float* dbu = xs  + (size_t)BL * D_INNER_;  // BL*XPROJ_N
  float* yb  = dbu + (size_t)BL * XPROJ_N;   // BL*D_INNER
  float* ag  = yb  + (size_t)BL * D_INNER_;  // BL*ACT_DIM
  float* xn  = ag  + (size_t)BL * ACT_DIM_;  // BL*HID

  const dim3 blk(256);
  auto gg = [](int N) { return dim3(((N >> 4) + 7) / 8, BL / 16); };

  // --- embed: x = latents @ lpw^T + lpb ; += a_emb[actions] @ apw^T + apb ; += pos
  gemm_f32_wmma<<<gg(HID_), blk, 0, stream>>>(latents, lpw, lpb, nullptr, x,
                                              BL, HID_, LATENT_);
  gather_actions<<<(BL*ACT_DIM_ + 255)/256, blk, 0, stream>>>(actions, aemb, ag);
  gemm_f32_wmma<<<gg(HID_), blk, 0, stream>>>(ag, apw, apb, x, x,
                                              BL, HID_, ACT_DIM_);
  add_pos<<<(BL*HID_ + 255)/256, blk, 0, stream>>>(x, pos);

  // --- 12 Mamba layers
  for (int l = 0; l < NLAYERS_; ++l) {
    gemm_f32_wmma<<<gg(2*D_INNER_), blk, 0, stream>>>(
        x, in_w + (size_t)l * 2*D_INNER_ * HID_, in_b + (size_t)l * 2*D_INNER_,
        nullptr, xz, BL, 2*D_INNER_, HID_);

    conv_silu<<<(B_*D_INNER_ + 255)/256, blk, 0, stream>>>(
        xz, conv_w + (size_t)l * D_INNER_ * D_CONV_, conv_b + (size_t)l * D_INNER_, xs);

    gemm_f32_wmma<<<gg(XPROJ_N), blk, 0, stream>>>(
        xs, xproj_w + (size_t)l * XPROJ_N * D_INNER_, xproj_b + (size_t)l * XPROJ_N,
        nullptr, dbu, BL, XPROJ_N, D_INNER_);

    ssm_scan<<<(B_*D_INNER_ + 255)/256, blk, 0, stream>>>(
        dbu, xz, dt_w + (size_t)l * D_INNER_ * D_STATE_, dt_b + (size_t)l * D_INNER_,
        A_log + (size_t)l * D_INNER_ * D_STATE_, Dp + (size_t)l * D_INNER_, yb);

    gemm_f32_wmma<<<gg(HID_), blk, 0, stream>>>(
        yb, blk_out_w + (size_t)l * HID_ * D_INNER_, blk_out_b + (size_t)l * HID_,
        x, x, BL, HID_, D_INNER_);
  }

  // --- final layernorm + output projection
  layernorm<<<BL, blk, 0, stream>>>(x, norm_g, norm_b, xn);
  gemm_f32_wmma<<<gg(LATENT_), blk, 0, stream>>>(xn, out_w, out_b, nullptr, out,
                                                 BL, LATENT_, HID_);
}